// StyleLoss_15599321219646
// MI455X (gfx1250) — compile-verified
//
#include <hip/hip_runtime.h>
#include <hip/hip_bf16.h>

typedef __attribute__((ext_vector_type(2))) float v2f;
typedef __attribute__((ext_vector_type(4))) float f4;
typedef __attribute__((ext_vector_type(8))) float v8f;

// ---------------------------------------------------------------------------
// Exact f32 wave32 sum via the matrix engine.
// A (16x4): VGPR0 = lane value, VGPR1 = 0  -> each lane's value appears exactly
// once in the A tile. B = all-ones (both VGPRs), so D[m][n] = rowsum_m(A).
// A lane holds one D column (8 VGPRs = rows 0-7 or 8-15); summing them and
// adding across the two half-waves yields the total in every lane.
// EXEC must be all ones at the call site (no divergence before this point).
// ---------------------------------------------------------------------------
__device__ __forceinline__ float wave_sum_wmma(float v) {
    v2f a; a.x = v;    a.y = 0.0f;
    v2f b; b.x = 1.0f; b.y = 1.0f;
    v8f c = {};
    v8f d = __builtin_amdgcn_wmma_f32_16x16x4_f32(
        /*neg_a=*/false, a, /*neg_b=*/false, b,
        /*c_mod=*/(short)0, c, /*reuse_a=*/false, /*reuse_b=*/false);
    float s = d[0] + d[1] + d[2] + d[3] + d[4] + d[5] + d[6] + d[7];
    s += __shfl_xor(s, 16, 32);   // combine rows 0-7 with rows 8-15
    return s;
}

__device__ __forceinline__ void acc5(float v, float* s) {
    float v2 = v * v, v3 = v2 * v, v4 = v2 * v2, v5 = v4 * v;
    s[0] += v; s[1] += v2; s[2] += v3; s[3] += v4; s[4] += v5;
}

// ---------------------------------------------------------------------------
// Kernel 1: one block per (b, c) pair. Streams 16384 contiguous floats of x
// and of target (fully coalesced float4 NT loads), accumulates raw moment
// partial sums S1..S5 for both, reduces per wave with WMMA, folds 8 waves via
// LDS, writes 10 floats per block to the workspace (deterministic, no atomics).
// ---------------------------------------------------------------------------
__global__ __launch_bounds__(256) void styleloss_moments_partial(
    const float* __restrict__ x, const float* __restrict__ y,
    float* __restrict__ part) {
    const int blk = blockIdx.x;          // blk = b*256 + c (matches memory layout)
    const int b   = blk >> 8;
    const int c   = blk & 255;
    const long base = (long)blk * 16384; // H*W contiguous floats per (b,c)
    const f4* __restrict__ xp = (const f4*)(x + base);
    const f4* __restrict__ yp = (const f4*)(y + base);
    const int t = threadIdx.x;

    float s[10] = {0.f,0.f,0.f,0.f,0.f,0.f,0.f,0.f,0.f,0.f};

    #pragma unroll
    for (int i = 0; i < 16; ++i) {
        f4 vx = __builtin_nontemporal_load(xp + i * 256 + t);
        f4 vy = __builtin_nontemporal_load(yp + i * 256 + t);
        acc5(vx.x, s);     acc5(vx.y, s);     acc5(vx.z, s);     acc5(vx.w, s);
        acc5(vy.x, s + 5); acc5(vy.y, s + 5); acc5(vy.z, s + 5); acc5(vy.w, s + 5);
    }

    // wave32 reduction on the matrix engine (all 256 threads active -> EXEC ok)
    #pragma unroll
    for (int j = 0; j < 10; ++j) s[j] = wave_sum_wmma(s[j]);

    __shared__ float lds[8 * 10];
    const int wave = t >> 5;
    if ((t & 31) == 0) {
        #pragma unroll
        for (int j = 0; j < 10; ++j) lds[wave * 10 + j] = s[j];
    }
    __syncthreads();

    if (t < 10) {
        float acc = 0.f;
        #pragma unroll
        for (int w = 0; w < 8; ++w) acc += lds[w * 10 + t];
        // layout: [channel][batch][moment] so the finalize kernel reads
        // a contiguous 80-float run per channel
        part[((long)(c * 8 + b)) * 10 + t] = acc;
    }
}

// raw moments r1..r5 -> mean + central moments mu2..mu5
__device__ __forceinline__ void central_moments(const float* S, float invn, float* mu) {
    float r1 = S[0]*invn, r2 = S[1]*invn, r3 = S[2]*invn, r4 = S[3]*invn, r5 = S[4]*invn;
    float m = r1, m2 = m*m, m3 = m2*m, m4 = m2*m2, m5 = m4*m;
    mu[0] = m;
    mu[1] = r2 - m2;
    mu[2] = r3 - 3.f*m*r2 + 2.f*m3;
    mu[3] = r4 - 4.f*m*r3 + 6.f*m2*r2 - 3.f*m4;
    mu[4] = r5 - 5.f*m*r4 + 10.f*m2*r3 - 10.f*m3*r2 + 4.f*m5;
}

// ---------------------------------------------------------------------------
// Kernel 2: one block, 256 threads (thread c == channel c). Folds the 8
// per-batch partials, converts to central moments, WMMA-reduces the 5
// squared-difference terms across all 256 channels, thread 0 emits the loss.
// ---------------------------------------------------------------------------
__global__ __launch_bounds__(256) void styleloss_finalize(
    const float* __restrict__ part, float* __restrict__ out) {
    const int c = threadIdx.x;

    float S[10] = {0.f,0.f,0.f,0.f,0.f,0.f,0.f,0.f,0.f,0.f};
    #pragma unroll
    for (int b = 0; b < 8; ++b) {
        #pragma unroll
        for (int j = 0; j < 10; ++j) S[j] += part[(long)(c * 8 + b) * 10 + j];
    }

    const float invn = 1.0f / 131072.0f;   // B*H*W = 8*128*128
    float mux[5], muy[5];
    central_moments(S,     invn, mux);
    central_moments(S + 5, invn, muy);

    float term[5];
    #pragma unroll
    for (int i = 0; i < 5; ++i) {
        float d = mux[i] - muy[i];
        term[i] = d * d;
    }

    // sum each term over 32 channels per wave on the matrix engine
    #pragma unroll
    for (int i = 0; i < 5; ++i) term[i] = wave_sum_wmma(term[i]);

    __shared__ float lds[8 * 5];
    const int wave = c >> 5;
    if ((c & 31) == 0) {
        #pragma unroll
        for (int i = 0; i < 5; ++i) lds[wave * 5 + i] = term[i];
    }
    __syncthreads();

    if (c == 0) {
        float loss = 0.f;
        #pragma unroll
        for (int i = 0; i < 5; ++i) {
            float u = 0.f;
            #pragma unroll
            for (int w = 0; w < 8; ++w) u += lds[w * 5 + i];
            loss += sqrtf(u);              // WEIGHTS are all 1.0
        }
        out[0] = loss;
    }
}

extern "C" void kernel_launch(void* const* d_in, const int* in_sizes, int n_in,
                              void* d_out, int out_size, void* d_ws, size_t ws_size,
                              hipStream_t stream) {
    const float* x = (const float*)d_in[0];
    const float* y = (const float*)d_in[1];
    float* part = (float*)d_ws;            // 2048 blocks * 10 floats = 80 KiB

    styleloss_moments_partial<<<2048, 256, 0, stream>>>(x, y, part);
    styleloss_finalize<<<1, 256, 0, stream>>>(part, (float*)d_out);
}